// WTA_with_lateral_inhibition_4629974745676
// MI455X (gfx1250) — compile-verified
//
#include <hip/hip_runtime.h>
#include <climits>

// WTA with lateral inhibition, shape (32, 1e6) f32, topk=5, radius=5.
// HBM-bound: ~256MB traffic -> ~11us at 23.3 TB/s. Two kernels:
//  1) fused streaming pass: zero `out`, per-chunk (max, first-argmax, min)
//  2) per-row top-k simulation over chunk maxima; re-scan only the <=2
//     chunks touched by each 2*radius suppression window.

typedef float v4f __attribute__((ext_vector_type(4)));

#define TPB 256
#define CHUNK 4096
#define MAXCHUNKS 512   // supports N up to 2M with CHUNK=4096 (we have 245)
#define MAXWIN 16

__device__ __forceinline__ void combineMax(float& v, int& idx, float ov, int oidx) {
  // max with first-occurrence (smaller index) tie-break — matches jnp.argmax
  if (ov > v || (ov == v && oidx < idx)) { v = ov; idx = oidx; }
}

__device__ __forceinline__ void waveReduceMaxIdx(float& v, int& idx) {
#pragma unroll
  for (int m = 16; m >= 1; m >>= 1) {         // wave32: masks 16..1
    float ov = __shfl_xor(v, m, 32);
    int   oi = __shfl_xor(idx, m, 32);
    combineMax(v, idx, ov, oi);
  }
}

__device__ __forceinline__ float waveReduceMin(float v) {
#pragma unroll
  for (int m = 16; m >= 1; m >>= 1) v = fminf(v, __shfl_xor(v, m, 32));
  return v;
}

// Block-wide (max, argmax) reduction; result broadcast to all threads.
__device__ __forceinline__ void blockReduceMaxIdx(float& v, int& idx,
                                                  float* swv, int* swi) {
  __syncthreads();                   // protect scratch reuse across calls
  waveReduceMaxIdx(v, idx);
  const int lane = threadIdx.x & 31, wid = threadIdx.x >> 5;
  if (lane == 0) { swv[wid] = v; swi[wid] = idx; }
  __syncthreads();
  if (threadIdx.x < 32) {
    const int nw = blockDim.x >> 5;
    float vv = (lane < nw) ? swv[lane] : -__builtin_inff();
    int   ii = (lane < nw) ? swi[lane] : INT_MAX;
    waveReduceMaxIdx(vv, ii);
    if (lane == 0) { swv[0] = vv; swi[0] = ii; }
  }
  __syncthreads();
  v = swv[0]; idx = swi[0];
}

// ---- Kernel 1: streaming pass. Zeros `out`, emits per-chunk stats. ----
__global__ void wta_chunk_stats(const float* __restrict__ x, float* __restrict__ out,
                                float* __restrict__ cmax, int* __restrict__ cidx,
                                float* __restrict__ cmin, int N, int nchunks) {
  __shared__ float swv[TPB / 32];
  __shared__ int   swi[TPB / 32];
  __shared__ float smn[TPB / 32];

  const int row = blockIdx.x / nchunks;
  const int c   = blockIdx.x - row * nchunks;
  const long long rb = (long long)row * N;
  const int s = c * CHUNK;
  const int e = min(s + CHUNK, N);
  const int e4 = s + (((e - s) >> 2) << 2);   // end of full float4 quads

  float vmax = -__builtin_inff(); int imax = INT_MAX;
  float vmin =  __builtin_inff();
  const v4f zero = {0.0f, 0.0f, 0.0f, 0.0f};

  // b128 non-temporal stream: read x, zero out, track max/first-argmax/min.
  for (int i = s + (int)threadIdx.x * 4; i < e4; i += TPB * 4) {
    v4f v = __builtin_nontemporal_load((const v4f*)(x + rb + i));
    __builtin_prefetch(x + rb + i + TPB * 4, 0, 0);   // global_prefetch_b8
    __builtin_nontemporal_store(zero, (v4f*)(out + rb + i));
    if (v.x > vmax) { vmax = v.x; imax = i; }
    if (v.y > vmax) { vmax = v.y; imax = i + 1; }
    if (v.z > vmax) { vmax = v.z; imax = i + 2; }
    if (v.w > vmax) { vmax = v.w; imax = i + 3; }
    vmin = fminf(vmin, fminf(fminf(v.x, v.y), fminf(v.z, v.w)));
  }
  // scalar tail (tail indices are all larger, so strict '>' keeps first max)
  for (int i = e4 + (int)threadIdx.x; i < e; i += TPB) {
    float v = x[rb + i];
    out[rb + i] = 0.0f;
    if (v > vmax) { vmax = v; imax = i; }
    vmin = fminf(vmin, v);
  }

  blockReduceMaxIdx(vmax, imax, swv, swi);

  vmin = waveReduceMin(vmin);
  const int lane = threadIdx.x & 31, wid = threadIdx.x >> 5;
  if (lane == 0) smn[wid] = vmin;
  __syncthreads();
  if (threadIdx.x == 0) {
    float m = smn[0];
#pragma unroll
    for (int w = 1; w < TPB / 32; ++w) m = fminf(m, smn[w]);
    cmax[blockIdx.x] = vmax;
    cidx[blockIdx.x] = imax;
    cmin[blockIdx.x] = m;
  }
}

// ---- Kernel 2: per-row iterative WTA over chunk maxima. ----
__global__ void wta_select(const float* __restrict__ x, float* __restrict__ out,
                           const float* __restrict__ cmaxg,
                           const int* __restrict__ cidxg,
                           const float* __restrict__ cming,
                           int N, int nchunks,
                           const int* __restrict__ topk_p,
                           const int* __restrict__ radius_p) {
  __shared__ float cmaxS[MAXCHUNKS];
  __shared__ int   cidxS[MAXCHUNKS];
  __shared__ float swv[TPB / 32];
  __shared__ int   swi[TPB / 32];
  __shared__ float smn[TPB / 32];
  __shared__ int   s_wlo[MAXWIN], s_whi[MAXWIN];
  __shared__ int   s_nw, s_c0, s_c1, s_do;
  __shared__ float s_minv;

  const int row = blockIdx.x;
  const long long rb = (long long)row * N;
  int topk = *topk_p; if (topk > MAXWIN) topk = MAXWIN;
  const int radius = *radius_p;

  // Load chunk stats to LDS, reduce row min.
  float vmin = __builtin_inff();
  for (int c = threadIdx.x; c < nchunks; c += TPB) {
    cmaxS[c] = cmaxg[row * nchunks + c];
    cidxS[c] = cidxg[row * nchunks + c];
    vmin = fminf(vmin, cming[row * nchunks + c]);
  }
  vmin = waveReduceMin(vmin);
  const int lane = threadIdx.x & 31, wid = threadIdx.x >> 5;
  if (lane == 0) smn[wid] = vmin;
  if (threadIdx.x == 0) s_nw = 0;
  __syncthreads();
  if (threadIdx.x == 0) {
    float m = smn[0];
#pragma unroll
    for (int w = 1; w < TPB / 32; ++w) m = fminf(m, smn[w]);
    s_minv = m;
  }
  __syncthreads();
  const float minv = s_minv;

  for (int t = 0; t < topk; ++t) {
    // Global argmax over chunk maxima (first-occurrence tie-break).
    float v = -__builtin_inff(); int idx = INT_MAX;
    for (int c = threadIdx.x; c < nchunks; c += TPB)
      combineMax(v, idx, cmaxS[c], cidxS[c]);
    blockReduceMaxIdx(v, idx, swv, swi);

    if (threadIdx.x == 0) {
      out[rb + idx] = 1.0f;
      if (idx >= radius) {               // reference: no suppression if idx<radius
        int lo = idx - radius;
        int hi = idx + radius; if (hi > N) hi = N;
        int k = s_nw; s_wlo[k] = lo; s_whi[k] = hi; s_nw = k + 1;
        s_c0 = lo / CHUNK; s_c1 = (hi - 1) / CHUNK; s_do = 1;
      } else {
        s_do = 0;                        // same idx wins again next iter (idempotent)
      }
    }
    __syncthreads();

    if (s_do) {
      const int c0 = s_c0, c1 = s_c1, nw = s_nw;
      for (int cc = c0; cc <= c1; ++cc) {     // at most 2 chunks (2r=10 << CHUNK)
        const int cs = cc * CHUNK, ce = min(cs + CHUNK, N);
        float cv = -__builtin_inff(); int ci = INT_MAX;
        for (int i = cs + (int)threadIdx.x; i < ce; i += TPB) {
          float xv = x[rb + i];
          for (int j = 0; j < nw; ++j)
            if (i >= s_wlo[j] && i < s_whi[j]) { xv = minv; break; }
          combineMax(cv, ci, xv, i);          // suppressed vals keep their index
        }
        blockReduceMaxIdx(cv, ci, swv, swi);
        if (threadIdx.x == 0) { cmaxS[cc] = cv; cidxS[cc] = ci; }
      }
    }
    __syncthreads();
  }
}

extern "C" void kernel_launch(void* const* d_in, const int* in_sizes, int n_in,
                              void* d_out, int out_size, void* d_ws, size_t ws_size,
                              hipStream_t stream) {
  (void)n_in; (void)out_size; (void)ws_size;
  const float* x        = (const float*)d_in[0];
  const int*   topk_p   = (const int*)d_in[1];
  const int*   radius_p = (const int*)d_in[2];
  float*       out      = (float*)d_out;

  const int B = 32;                      // reference shape (32, 1000000)
  const int N = in_sizes[0] / B;
  const int nchunks = (N + CHUNK - 1) / CHUNK;   // 245 for N=1e6

  // Workspace: per (row, chunk) max/argmax/min  -> 3*32*245*4 = ~94 KB.
  float* cmax = (float*)d_ws;
  int*   cidx = (int*)(cmax + (size_t)B * nchunks);
  float* cmin = (float*)(cidx + (size_t)B * nchunks);

  wta_chunk_stats<<<B * nchunks, TPB, 0, stream>>>(x, out, cmax, cidx, cmin, N, nchunks);
  wta_select<<<B, TPB, 0, stream>>>(x, out, cmax, cidx, cmin, N, nchunks, topk_p, radius_p);
}